// LambdaConvLayer_49082886259416
// MI455X (gfx1250) — compile-verified
//
#include <hip/hip_runtime.h>
#include <math.h>

// Problem constants (from reference)
#define BB   8
#define CIN  512
#define CCTX 512
#define KD   64
#define VD   256
#define TT   1024
#define KS   23
#define EPS  1e-05f

typedef __attribute__((ext_vector_type(2))) float v2f;
typedef __attribute__((ext_vector_type(8))) float v8f;

__device__ __forceinline__ v8f wmma_f32_4(v2f a, v2f b, v8f c) {
  // V_WMMA_F32_16X16X4_F32 : D(16x16 f32) = A(16x4 f32) x B(4x16 f32) + C
  return __builtin_amdgcn_wmma_f32_16x16x4_f32(false, a, false, b, (short)0, c,
                                               false, false);
}

// ---------------------------------------------------------------------------
// Kernel 1: projection GEMM  P[b] (M x T) = W (M x C) @ X[b] (C x T) + bias
// One wave computes one 16x16 tile of P. ISA lane layout for f32 WMMA:
//   A (16x4):  lane<16 -> row=lane, K={0,1}; lane>=16 -> row=lane-16, K={2,3}
//   B (4x16):  lane<16 -> col=lane, VGPR0=K0,VGPR1=K1; lane>=16 -> K2,K3
//   C (16x16): vgpr j -> row = j + 8*(lane>=16), col = lane&15
// ---------------------------------------------------------------------------
__global__ __launch_bounds__(32) void proj_gemm_kernel(
    const float* __restrict__ W, const float* __restrict__ bias,
    const float* __restrict__ X, float* __restrict__ P, int M, int C) {
  const int t0   = blockIdx.x * 16;
  const int m0   = blockIdx.y * 16;
  const int b    = blockIdx.z;
  const int lane = threadIdx.x;
  const int half = lane >> 4;   // 0 or 1
  const int l16  = lane & 15;

  const float* Xb = X + (size_t)b * C * TT;
  v8f acc = {};
  for (int k0 = 0; k0 < C; k0 += 4) {
    const float* pa = W + (size_t)(m0 + l16) * C + (k0 + half * 2);
    v2f a = { pa[0], pa[1] };
    const float* pb = Xb + (size_t)(k0 + half * 2) * TT + (t0 + l16);
    v2f bm = { pb[0], pb[TT] };
    acc = wmma_f32_4(a, bm, acc);
  }
  float* Pb = P + (size_t)b * M * TT;
#pragma unroll
  for (int j = 0; j < 8; ++j) {
    const int row = m0 + j + half * 8;
    Pb[(size_t)row * TT + t0 + l16] = acc[j] + bias[row];
  }
}

// ---------------------------------------------------------------------------
// Kernel 2: softmax over T for each (b,k) row of keys (in place)
// ---------------------------------------------------------------------------
__global__ __launch_bounds__(256) void softmax_rows_kernel(float* __restrict__ keys) {
  __shared__ float red[256];
  const int tid = threadIdx.x;
  float* p = keys + (size_t)blockIdx.x * TT;

  float m = -INFINITY;
  for (int i = tid; i < TT; i += 256) m = fmaxf(m, p[i]);
  red[tid] = m;
  __syncthreads();
  for (int s = 128; s > 0; s >>= 1) {
    if (tid < s) red[tid] = fmaxf(red[tid], red[tid + s]);
    __syncthreads();
  }
  m = red[0];
  __syncthreads();

  float sum = 0.0f;
  for (int i = tid; i < TT; i += 256) {
    float e = expf(p[i] - m);
    p[i] = e;
    sum += e;
  }
  red[tid] = sum;
  __syncthreads();
  for (int s = 128; s > 0; s >>= 1) {
    if (tid < s) red[tid] += red[tid + s];
    __syncthreads();
  }
  const float inv = 1.0f / red[0];
  for (int i = tid; i < TT; i += 256) p[i] *= inv;
}

// ---------------------------------------------------------------------------
// Kernel 3: BatchNorm stats over (b, t) per channel -> scale/shift
//   scale[c] = gamma[c] * rsqrt(var + eps);  shift[c] = beta[c] - mean*scale
// data layout: (BB, Cch, TT). One block per channel.
// ---------------------------------------------------------------------------
__global__ __launch_bounds__(256) void bn_stats_kernel(
    const float* __restrict__ data, const float* __restrict__ gamma,
    const float* __restrict__ beta, float* __restrict__ scale,
    float* __restrict__ shift, int Cch) {
  __shared__ float s_sum[256];
  __shared__ float s_sq[256];
  const int c = blockIdx.x;
  const int tid = threadIdx.x;
  float sum = 0.0f, sq = 0.0f;
  for (int b = 0; b < BB; ++b) {
    const float* p = data + ((size_t)b * Cch + c) * TT;
    for (int i = tid; i < TT; i += 256) {
      float v = p[i];
      sum += v;
      sq += v * v;
    }
  }
  s_sum[tid] = sum;
  s_sq[tid] = sq;
  __syncthreads();
  for (int s = 128; s > 0; s >>= 1) {
    if (tid < s) {
      s_sum[tid] += s_sum[tid + s];
      s_sq[tid] += s_sq[tid + s];
    }
    __syncthreads();
  }
  if (tid == 0) {
    const float n = (float)(BB * TT);
    const float mean = s_sum[0] / n;
    const float var = s_sq[0] / n - mean * mean;
    const float sc = gamma[c] * rsqrtf(var + EPS);
    scale[c] = sc;
    shift[c] = beta[c] - mean * sc;
  }
}

// ---------------------------------------------------------------------------
// Kernel 4: apply per-channel affine in place. layout (BB, Cch, TT)
// ---------------------------------------------------------------------------
__global__ __launch_bounds__(256) void bn_apply_kernel(
    float* __restrict__ data, const float* __restrict__ scale,
    const float* __restrict__ shift, int Cch, int n) {
  const int idx = blockIdx.x * 256 + threadIdx.x;
  if (idx >= n) return;
  const int c = (idx >> 10) % Cch;  // idx / TT (TT==1024)
  data[idx] = data[idx] * scale[c] + shift[c];
}

// ---------------------------------------------------------------------------
// Kernel 5: content lambda  L[b](64x256): L[k][v] = sum_t NK[b][k][t]*Vn[b][v][t]
// A = NK (64 x T) row-major; B[t][v] = Vn[v][t] so B loads are float2 along t.
// ---------------------------------------------------------------------------
__global__ __launch_bounds__(32) void lambda_gemm_kernel(
    const float* __restrict__ NK, const float* __restrict__ Vn,
    float* __restrict__ L) {
  const int n0   = blockIdx.x * 16;  // v
  const int m0   = blockIdx.y * 16;  // k
  const int b    = blockIdx.z;
  const int lane = threadIdx.x;
  const int half = lane >> 4;
  const int l16  = lane & 15;

  const float* A  = NK + (size_t)b * KD * TT;
  const float* Bv = Vn + (size_t)b * VD * TT;
  v8f acc = {};
  for (int t0 = 0; t0 < TT; t0 += 4) {
    const float* pa = A + (size_t)(m0 + l16) * TT + (t0 + half * 2);
    v2f a = { pa[0], pa[1] };
    const float* pb = Bv + (size_t)(n0 + l16) * TT + (t0 + half * 2);
    v2f bm = { pb[0], pb[1] };
    acc = wmma_f32_4(a, bm, acc);
  }
  float* Lb = L + (size_t)b * KD * VD;
#pragma unroll
  for (int j = 0; j < 8; ++j) {
    const int row = m0 + j + half * 8;  // k index
    Lb[(size_t)row * VD + n0 + l16] = acc[j];
  }
}

// ---------------------------------------------------------------------------
// Kernel 6: qp[b][t][s] = sum_k Qn[b][k][t]*pos_w[k][s] (s<23); slot 23 = pos_b dot
// ---------------------------------------------------------------------------
__global__ __launch_bounds__(256) void qp_kernel(
    const float* __restrict__ Qn, const float* __restrict__ pos_w,
    const float* __restrict__ pos_b, float* __restrict__ qp) {
  const int idx = blockIdx.x * 256 + threadIdx.x;  // b*TT + t
  if (idx >= BB * TT) return;
  const int b = idx >> 10;
  const int t = idx & (TT - 1);
  const float* Qb = Qn + (size_t)b * KD * TT + t;
  float acc[24];
#pragma unroll
  for (int s = 0; s < 24; ++s) acc[s] = 0.0f;
  for (int k = 0; k < KD; ++k) {
    const float q = Qb[(size_t)k * TT];
    const float* pw = pos_w + k * KS;
#pragma unroll
    for (int s = 0; s < KS; ++s) acc[s] += q * pw[s];
    acc[23] += q * pos_b[k];
  }
  float* o = qp + (size_t)idx * 24;
#pragma unroll
  for (int s = 0; s < 24; ++s) o[s] = acc[s];
}

// ---------------------------------------------------------------------------
// Kernel 7: final output (B, V, T):
//   out[b][v][t] = sum_k L[b][k][v]*Qn[b][k][t]                (WMMA)
//                + sum_s qp[b][t][s]*Vn[b][v][t+s-11] + qp[b][t][23]  (stencil)
// A[m][kk] = L[kk][v0+m] (strided), B = Qn rows.
// ---------------------------------------------------------------------------
__global__ __launch_bounds__(32) void out_kernel(
    const float* __restrict__ L, const float* __restrict__ Qn,
    const float* __restrict__ Vn, const float* __restrict__ qp,
    float* __restrict__ out) {
  const int t0   = blockIdx.x * 16;
  const int v0   = blockIdx.y * 16;
  const int b    = blockIdx.z;
  const int lane = threadIdx.x;
  const int half = lane >> 4;
  const int l16  = lane & 15;

  const float* Lb = L + (size_t)b * KD * VD;
  const float* Qb = Qn + (size_t)b * KD * TT;
  v8f acc = {};
#pragma unroll
  for (int k0 = 0; k0 < KD; k0 += 4) {
    const float* pa = Lb + (size_t)(k0 + half * 2) * VD + (v0 + l16);
    v2f a = { pa[0], pa[VD] };
    const float* pb = Qb + (size_t)(k0 + half * 2) * TT + (t0 + l16);
    v2f bm = { pb[0], pb[TT] };
    acc = wmma_f32_4(a, bm, acc);
  }

  const int t = t0 + l16;  // output column is the same for both lane halves
  const float* qpt = qp + ((size_t)b * TT + t) * 24;
  float qpl[24];
#pragma unroll
  for (int s = 0; s < 24; ++s) qpl[s] = qpt[s];

  const float* Vb = Vn + (size_t)b * VD * TT;
  float* Ob = out + (size_t)b * VD * TT;
#pragma unroll
  for (int j = 0; j < 8; ++j) {
    const int v = v0 + j + half * 8;
    const float* Vrow = Vb + (size_t)v * TT;
    float r = acc[j] + qpl[23];
#pragma unroll
    for (int s = 0; s < KS; ++s) {
      const int tt = t + s - (KS / 2);
      const float val = (tt >= 0 && tt < TT) ? Vrow[tt] : 0.0f;
      r += qpl[s] * val;
    }
    Ob[(size_t)v * TT + t] = r;
  }
}

// ---------------------------------------------------------------------------
// Host side
// ---------------------------------------------------------------------------
extern "C" void kernel_launch(void* const* d_in, const int* in_sizes, int n_in,
                              void* d_out, int out_size, void* d_ws, size_t ws_size,
                              hipStream_t stream) {
  const float* x       = (const float*)d_in[0];
  const float* context = (const float*)d_in[1];
  const float* Wq      = (const float*)d_in[2];
  const float* bq      = (const float*)d_in[3];
  const float* Wk      = (const float*)d_in[4];
  const float* bk      = (const float*)d_in[5];
  const float* Wv      = (const float*)d_in[6];
  const float* bv      = (const float*)d_in[7];
  const float* gamma_q = (const float*)d_in[8];
  const float* beta_q  = (const float*)d_in[9];
  const float* gamma_v = (const float*)d_in[10];
  const float* beta_v  = (const float*)d_in[11];
  const float* pos_w   = (const float*)d_in[12];
  const float* pos_b   = (const float*)d_in[13];
  float* out = (float*)d_out;

  // Workspace carve-up (floats)
  const size_t N_Q  = (size_t)BB * KD * TT;  // 524288
  const size_t N_K  = (size_t)BB * KD * TT;
  const size_t N_V  = (size_t)BB * VD * TT;  // 2097152
  const size_t N_L  = (size_t)BB * KD * VD;  // 131072
  const size_t N_QP = (size_t)BB * TT * 24;  // 196608
  float* ws   = (float*)d_ws;
  float* q    = ws;
  float* keys = q + N_Q;
  float* vals = keys + N_K;
  float* lam  = vals + N_V;
  float* qp   = lam + N_L;
  float* sc_q = qp + N_QP;
  float* sh_q = sc_q + KD;
  float* sc_v = sh_q + KD;
  float* sh_v = sc_v + VD;

  // 1) projections (WMMA f32)
  proj_gemm_kernel<<<dim3(TT / 16, KD / 16, BB), 32, 0, stream>>>(Wq, bq, x, q, KD, CIN);
  proj_gemm_kernel<<<dim3(TT / 16, KD / 16, BB), 32, 0, stream>>>(Wk, bk, context, keys, KD, CCTX);
  proj_gemm_kernel<<<dim3(TT / 16, VD / 16, BB), 32, 0, stream>>>(Wv, bv, context, vals, VD, CCTX);

  // 2) softmax over T for keys
  softmax_rows_kernel<<<BB * KD, 256, 0, stream>>>(keys);

  // 3) BN stats + apply (in place) for q and vals
  bn_stats_kernel<<<KD, 256, 0, stream>>>(q, gamma_q, beta_q, sc_q, sh_q, KD);
  bn_stats_kernel<<<VD, 256, 0, stream>>>(vals, gamma_v, beta_v, sc_v, sh_v, VD);
  bn_apply_kernel<<<(int)((N_Q + 255) / 256), 256, 0, stream>>>(q, sc_q, sh_q, KD, (int)N_Q);
  bn_apply_kernel<<<(int)((N_V + 255) / 256), 256, 0, stream>>>(vals, sc_v, sh_v, VD, (int)N_V);

  // 4) content lambda (WMMA f32)
  lambda_gemm_kernel<<<dim3(VD / 16, KD / 16, BB), 32, 0, stream>>>(keys, vals, lam);

  // 5) qp = Qn^T @ pos_w (+ pos_b dot)
  qp_kernel<<<(BB * TT) / 256, 256, 0, stream>>>(q, pos_w, pos_b, qp);

  // 6) fused content_out + positional_out -> out (B, V, T)
  out_kernel<<<dim3(TT / 16, VD / 16, BB), 32, 0, stream>>>(lam, q, vals, qp, out);
}